// ShapeGraphEmbedder_79413945303069
// MI455X (gfx1250) — compile-verified
//
#include <hip/hip_runtime.h>
#include <hip/hip_bf16.h>
#include <math.h>

#define N_NODES   102400
#define G_GRAPHS  1024
#define E_EDGES   819200
#define IN_DIM    96
#define GNN_DIM   64
#define LATENT    128
#define OUT_COLS  288
#define QKVS_COLS 256
#define V_ATOM    100
#define ATT_SCALE 0.125f   // 1/sqrt(64)

typedef float v2f __attribute__((ext_vector_type(2)));
typedef float v8f __attribute__((ext_vector_type(8)));

// ---- order-preserving float <-> uint encoding for atomic max over floats ----
__device__ __forceinline__ unsigned encf(float f) {
  unsigned u = __float_as_uint(f);
  return (u & 0x80000000u) ? ~u : (u | 0x80000000u);
}
__device__ __forceinline__ float decf(unsigned k) {
  unsigned u = (k & 0x80000000u) ? (k & 0x7FFFFFFFu) : ~k;
  return __uint_as_float(u);
}

// ---- x0 = concat(shape_id_table[idx], shape_mult_table[mult]) -> d_out[:, 0:96] ----
__global__ void embed_kernel(const int* __restrict__ sidx, const int* __restrict__ smult,
                             const float* __restrict__ idt, const float* __restrict__ mt,
                             float* __restrict__ out) {
  int i = blockIdx.x * blockDim.x + threadIdx.x;
  if (i >= N_NODES * IN_DIM) return;
  int n = i / IN_DIM, c = i - n * IN_DIM;
  float v = (c < 64) ? idt[sidx[n] * 64 + c] : mt[smult[n] * 32 + (c - 64)];
  out[(size_t)n * OUT_COLS + c] = v;
}

// ---- serial prefix sum of num_nodes_hgraph (G=1024, trivial) ----
__global__ void prefix_kernel(const int* __restrict__ nn, int* __restrict__ offs) {
  if (blockIdx.x == 0 && threadIdx.x == 0) {
    int acc = 0; offs[0] = 0;
    for (int g = 0; g < G_GRAPHS; ++g) { acc += nn[g]; offs[g + 1] = acc; }
  }
}

// ---- z_rep -> d_out[:, 160:288] via binary search on offsets ----
__global__ void zfill_kernel(const float* __restrict__ z, const int* __restrict__ offs,
                             float* __restrict__ out) {
  int i = blockIdx.x * blockDim.x + threadIdx.x;
  if (i >= N_NODES * LATENT) return;
  int n = i >> 7, c = i & 127;
  int lo = 0, hi = G_GRAPHS;
  while (lo + 1 < hi) { int mid = (lo + hi) >> 1; if (offs[mid] <= n) lo = mid; else hi = mid; }
  out[(size_t)n * OUT_COLS + 160 + c] = z[lo * LATENT + c];
}

// ---- pack Wq|Wk|Wv|Wskip into (dK x 256) + bias(256) ----
__global__ void pack_w_kernel(const float* __restrict__ Wq, const float* __restrict__ bq,
                              const float* __restrict__ Wk, const float* __restrict__ bk,
                              const float* __restrict__ Wv, const float* __restrict__ bv,
                              const float* __restrict__ Ws, const float* __restrict__ bs,
                              int dK, float* __restrict__ Wcat, float* __restrict__ bias) {
  int i = blockIdx.x * blockDim.x + threadIdx.x;
  if (i >= dK * QKVS_COLS) return;
  int k = i >> 8, c = i & 255;
  float w;
  if (c < 64)       w = Wq[k * 64 + c];
  else if (c < 128) w = Wk[k * 64 + (c - 64)];
  else if (c < 192) w = Wv[k * 64 + (c - 128)];
  else              w = Ws[k * 64 + (c - 192)];
  Wcat[i] = w;
  if (k == 0) {
    float b;
    if (c < 64)       b = bq[c];
    else if (c < 128) b = bk[c - 64];
    else if (c < 192) b = bv[c - 128];
    else              b = bs[c - 192];
    bias[c] = b;
  }
}

// ---- T = atom_id_table @ We  (100 x 64, tiny) ----
__global__ void edge_table_kernel(const float* __restrict__ atom, const float* __restrict__ We,
                                  float* __restrict__ T) {
  int i = blockIdx.x * blockDim.x + threadIdx.x;
  if (i >= V_ATOM * 64) return;
  int v = i >> 6, c = i & 63;
  float s = 0.f;
  #pragma unroll 8
  for (int k = 0; k < 64; ++k) s += atom[v * 64 + k] * We[k * 64 + c];
  T[i] = s;
}

// ---- QKVS = A(dK) @ Wcat(dK x 256) + bias, via V_WMMA_F32_16X16X4_F32 ----
// block = 512 threads (16 waves); each wave owns one 16x16 output tile of a
// 16-row x 256-col panel. N % 16 == 0, dK % 4 == 0 -> no guards, EXEC all ones.
__global__ void gemm_qkvs_wmma(const float* __restrict__ A, int lda, int dK,
                               const float* __restrict__ W, const float* __restrict__ bias,
                               float* __restrict__ C) {
  const int lane = threadIdx.x & 31;
  const int wave = threadIdx.x >> 5;        // 0..15 -> output column tile
  const int row0 = blockIdx.x << 4;
  const int col0 = wave << 4;
  const int hl   = lane >> 4;               // lane-half selects K pair (0,1) vs (2,3)
  const int m    = lane & 15;

  v8f acc = {0.f, 0.f, 0.f, 0.f, 0.f, 0.f, 0.f, 0.f};
  const float* Arow = A + (size_t)(row0 + m) * lda;
  for (int k0 = 0; k0 < dK; k0 += 4) {
    v2f a, b;
    a.x = Arow[k0 + 2 * hl];
    a.y = Arow[k0 + 2 * hl + 1];
    b.x = W[(k0 + 2 * hl) * QKVS_COLS + col0 + m];
    b.y = W[(k0 + 2 * hl + 1) * QKVS_COLS + col0 + m];
    acc = __builtin_amdgcn_wmma_f32_16x16x4_f32(false, a, false, b, (short)0, acc,
                                                false, false);
  }
  const int col = col0 + m;
  const float bv = bias[col];
  #pragma unroll
  for (int r = 0; r < 8; ++r) {
    int mr = r + 8 * hl;                    // C layout: vgpr r, lane<16 -> M=r, else M=r+8
    C[(size_t)(row0 + mr) * QKVS_COLS + col] = acc[r] + bv;
  }
}

// ---- pass 1: logits per edge + segment max (one wave per edge) ----
__global__ void edge_logits_kernel(const int* __restrict__ src, const int* __restrict__ dst,
                                   const int* __restrict__ join, const float* __restrict__ QKVS,
                                   const float* __restrict__ T, float* __restrict__ logits,
                                   unsigned* __restrict__ mkey) {
  int gw = (blockIdx.x * blockDim.x + threadIdx.x) >> 5;
  int lane = threadIdx.x & 31;
  if (gw >= E_EDGES) return;
  int s = src[gw], d = dst[gw], j = join[gw] - 1;
  const float2 q  = *(const float2*)&QKVS[(size_t)d * QKVS_COLS + 2 * lane];
  const float2 kk = *(const float2*)&QKVS[(size_t)s * QKVS_COLS + 64 + 2 * lane];
  const float2 tt = *(const float2*)&T[j * 64 + 2 * lane];
  float p = q.x * (kk.x + tt.x) + q.y * (kk.y + tt.y);
  p += __shfl_down(p, 16); p += __shfl_down(p, 8); p += __shfl_down(p, 4);
  p += __shfl_down(p, 2);  p += __shfl_down(p, 1);
  if (lane == 0) {
    float lg = p * ATT_SCALE;
    logits[gw] = lg;
    atomicMax(&mkey[d], encf(lg));
  }
}

// ---- pass 2: a = exp(logit - max[dst]); denom[dst] += a ----
__global__ void edge_exp_kernel(const int* __restrict__ dst, const float* __restrict__ logits,
                                const unsigned* __restrict__ mkey, float* __restrict__ aexp,
                                float* __restrict__ denom) {
  int e = blockIdx.x * blockDim.x + threadIdx.x;
  if (e >= E_EDGES) return;
  int d = dst[e];
  float a = expf(logits[e] - decf(mkey[d]));
  aexp[e] = a;
  unsafeAtomicAdd(&denom[d], a);
}

// ---- pass 3: out[dst] += alpha * (v[src] + T[join])  (one wave per edge) ----
__global__ void edge_msg_kernel(const int* __restrict__ src, const int* __restrict__ dst,
                                const int* __restrict__ join, const float* __restrict__ QKVS,
                                const float* __restrict__ T, const float* __restrict__ aexp,
                                const float* __restrict__ denom, float* __restrict__ hout) {
  int gw = (blockIdx.x * blockDim.x + threadIdx.x) >> 5;
  int lane = threadIdx.x & 31;
  if (gw >= E_EDGES) return;
  int s = src[gw], d = dst[gw], j = join[gw] - 1;
  float alpha = aexp[gw] / denom[d];
  const float2 vv = *(const float2*)&QKVS[(size_t)s * QKVS_COLS + 128 + 2 * lane];
  const float2 tt = *(const float2*)&T[j * 64 + 2 * lane];
  float* o = &hout[(size_t)d * GNN_DIM + 2 * lane];
  unsafeAtomicAdd(o,     alpha * (vv.x + tt.x));
  unsafeAtomicAdd(o + 1, alpha * (vv.y + tt.y));
}

// ---- h = relu(msg_sum + skip); last layer also writes d_out[:, 96:160] ----
__global__ void finalize_kernel(const float* __restrict__ QKVS, float* __restrict__ hout,
                                float* __restrict__ out_final, int last) {
  int i = blockIdx.x * blockDim.x + threadIdx.x;
  if (i >= N_NODES * GNN_DIM) return;
  int n = i >> 6, c = i & 63;
  float v = hout[i] + QKVS[(size_t)n * QKVS_COLS + 192 + c];
  v = fmaxf(v, 0.f);
  hout[i] = v;
  if (last) out_final[(size_t)n * OUT_COLS + 96 + c] = v;
}

extern "C" void kernel_launch(void* const* d_in, const int* in_sizes, int n_in,
                              void* d_out, int out_size, void* d_ws, size_t ws_size,
                              hipStream_t stream) {
  const int*   sidx  = (const int*)d_in[0];
  const int*   smult = (const int*)d_in[1];
  const int*   eidx  = (const int*)d_in[2];   // (2, E)
  const int*   join  = (const int*)d_in[3];
  const int*   nnh   = (const int*)d_in[4];
  const float* zg    = (const float*)d_in[5];
  const float* idt   = (const float*)d_in[6];
  const float* mt    = (const float*)d_in[7];
  const float* atom  = (const float*)d_in[8];
  const int* src = eidx;
  const int* dst = eidx + E_EDGES;
  float* out = (float*)d_out;

  // ---- workspace carve-up ----
  char* p = (char*)d_ws;
  float*    QKVS   = (float*)p;    p += (size_t)N_NODES * QKVS_COLS * 4;
  float*    hA     = (float*)p;    p += (size_t)N_NODES * GNN_DIM * 4;
  float*    hB     = (float*)p;    p += (size_t)N_NODES * GNN_DIM * 4;
  float*    logits = (float*)p;    p += (size_t)E_EDGES * 4;
  float*    aexp   = (float*)p;    p += (size_t)E_EDGES * 4;
  unsigned* mkey   = (unsigned*)p; p += (size_t)N_NODES * 4;
  float*    denom  = (float*)p;    p += (size_t)N_NODES * 4;
  float*    Wcat   = (float*)p;    p += (size_t)IN_DIM * QKVS_COLS * 4;
  float*    biasv  = (float*)p;    p += 256 * 4;
  float*    T      = (float*)p;    p += (size_t)V_ATOM * 64 * 4;
  int*      offs   = (int*)p;      p += (G_GRAPHS + 1) * 4;

  // ---- static parts of the output ----
  embed_kernel<<<(N_NODES * IN_DIM + 255) / 256, 256, 0, stream>>>(sidx, smult, idt, mt, out);
  prefix_kernel<<<1, 1, 0, stream>>>(nnh, offs);
  zfill_kernel<<<(N_NODES * LATENT + 255) / 256, 256, 0, stream>>>(zg, offs, out);

  // ---- 3 GNN layers ----
  for (int l = 0; l < 3; ++l) {
    const float* Wq = (const float*)d_in[9 + l * 9 + 0];
    const float* bq = (const float*)d_in[9 + l * 9 + 1];
    const float* Wk = (const float*)d_in[9 + l * 9 + 2];
    const float* bk = (const float*)d_in[9 + l * 9 + 3];
    const float* Wv = (const float*)d_in[9 + l * 9 + 4];
    const float* bv = (const float*)d_in[9 + l * 9 + 5];
    const float* We = (const float*)d_in[9 + l * 9 + 6];
    const float* Ws = (const float*)d_in[9 + l * 9 + 7];
    const float* bs = (const float*)d_in[9 + l * 9 + 8];

    const int dK  = (l == 0) ? IN_DIM : GNN_DIM;
    const int lda = (l == 0) ? OUT_COLS : GNN_DIM;
    const float* hin = (l == 0) ? out : ((l == 1) ? hA : hB);
    float* accum     = (l == 0) ? hA  : ((l == 1) ? hB : hA);

    pack_w_kernel<<<(dK * QKVS_COLS + 255) / 256, 256, 0, stream>>>(Wq, bq, Wk, bk, Wv, bv,
                                                                    Ws, bs, dK, Wcat, biasv);
    edge_table_kernel<<<(V_ATOM * 64 + 255) / 256, 256, 0, stream>>>(atom, We, T);
    hipMemsetAsync(mkey,  0, (size_t)N_NODES * 4, stream);
    hipMemsetAsync(denom, 0, (size_t)N_NODES * 4, stream);
    hipMemsetAsync(accum, 0, (size_t)N_NODES * GNN_DIM * 4, stream);

    gemm_qkvs_wmma<<<N_NODES / 16, 512, 0, stream>>>(hin, lda, dK, Wcat, biasv, QKVS);

    edge_logits_kernel<<<(E_EDGES * 32) / 256, 256, 0, stream>>>(src, dst, join, QKVS, T,
                                                                 logits, mkey);
    edge_exp_kernel<<<E_EDGES / 256, 256, 0, stream>>>(dst, logits, mkey, aexp, denom);
    edge_msg_kernel<<<(E_EDGES * 32) / 256, 256, 0, stream>>>(src, dst, join, QKVS, T,
                                                              aexp, denom, accum);
    finalize_kernel<<<(N_NODES * GNN_DIM) / 256, 256, 0, stream>>>(QKVS, accum, out, l == 2);
  }
}